// Graph_conv_layer_5188320494189
// MI455X (gfx1250) — compile-verified
//
#include <hip/hip_runtime.h>

// ---------------------------------------------------------------------------
// GCN layer: out = segment_sum_row( adj_val * (X@W)[adj_col] ) + bias
//   X: [100000 x 256] f32, W: [256 x 128] f32, 1.6M COO edges, bias: [128]
// Pipeline:
//   1) gemm_wmma:   support = X @ W   (f32 WMMA 16x16x4; X tile staged in LDS
//                                      by the Tensor Data Mover with HW row pad)
//   2) init_bias:   out[n, f] = bias[f]
//   3) spmm_atomic: out[row] += val * support[col]   (hw f32 atomics)
// ---------------------------------------------------------------------------

#define IN_F  256
#define OUT_F 128
#define LDA   260   // LDS row stride in floats: 260 % 64 == 4 -> conflict-free

typedef __attribute__((ext_vector_type(2))) float        v2f;
typedef __attribute__((ext_vector_type(8))) float        v8f;
typedef __attribute__((ext_vector_type(4))) unsigned int u32x4;
typedef __attribute__((ext_vector_type(8))) int          i32x8;
typedef __attribute__((ext_vector_type(4))) int          i32x4;

#if __has_builtin(__builtin_amdgcn_tensor_load_to_lds) && \
    __has_builtin(__builtin_amdgcn_s_wait_tensorcnt)
#define HAVE_TDM 1
#else
#define HAVE_TDM 0
#endif

// --- Kernel 1: support = X @ W via V_WMMA_F32_16X16X4_F32 -------------------
// Block: 256 threads (8 waves). Block tile: 16 rows x 128 cols.
// Wave w computes the 16x16 tile at columns [16w, 16w+16).
__global__ void __launch_bounds__(256)
gemm_wmma(const float* __restrict__ X, const float* __restrict__ W,
          float* __restrict__ S, int n_rows) {
    __shared__ float As[16 * LDA];

    const int m0  = blockIdx.x * 16;
    const int tid = threadIdx.x;

#if HAVE_TDM
    // Stage the 16x256 f32 X panel into LDS via the Tensor Data Mover.
    // HW padding: after every 256 DWORDs (one row) insert 4 DWORDs -> LDA=260.
    if (tid < 32) {                       // one wave issues the DMA descriptor
        unsigned long long ga  = (unsigned long long)(X + (size_t)m0 * IN_F);
        unsigned int       lds = (unsigned int)(unsigned long long)(void*)&As[0];

        u32x4 g0;
        g0[0] = 1u;                                   // count=1 (valid), user mode
        g0[1] = lds;                                  // lds_addr (bytes)
        g0[2] = (unsigned int)(ga & 0xFFFFFFFFu);     // global_addr[31:0]
        g0[3] = (unsigned int)((ga >> 32) & 0x01FFFFFFu)
              | (2u << 30);                           // global_addr[56:32] | type=2

        i32x8 g1;
        g1[0] = (2 << 16)     // data_size = 4 bytes
              | (1 << 20)     // pad_enable
              | (7 << 22)     // pad_interval: 256 DWORDs (one 256-float row)
              | (3 << 25);    // pad_amount: 4 DWORDs (16 B)
        g1[1] = (IN_F & 0xFFFF) << 16;                           // tensor_dim0 lo16
        g1[2] = ((n_rows & 0xFFFF) << 16);                       // dim0 hi16=0 | tensor_dim1 lo16
        g1[3] = ((n_rows >> 16) & 0xFFFF) | (IN_F << 16);        // tensor_dim1 hi16 | tile_dim0=256
        g1[4] = 16;                                              // tile_dim1=16 rows | tile_dim2=0
        g1[5] = IN_F;                                            // tensor_dim0_stride = 256
        g1[6] = 0;
        g1[7] = 0;
        i32x4 gz4 = {0, 0, 0, 0};                                // groups 2/3 unused (2D)
        i32x8 gz8 = {0, 0, 0, 0, 0, 0, 0, 0};

        // 6-arg form (clang-23 / therock-10.0 headers)
        __builtin_amdgcn_tensor_load_to_lds(g0, g1, gz4, gz4, gz8, 0);
        __builtin_amdgcn_s_wait_tensorcnt((short)0);   // s_wait_tensorcnt 0
    }
#else
    // Fallback: cooperative load of the 16x256 X panel (float4 per thread x4).
    {
        const float4* Xv = (const float4*)(X + (size_t)m0 * IN_F);
        #pragma unroll
        for (int i = 0; i < 4; ++i) {
            int   f4 = tid + i * 256;          // 1024 float4s total
            float4 v = Xv[f4];
            int r = (f4 * 4) / IN_F;
            int c = (f4 * 4) % IN_F;
            float* dst = &As[r * LDA + c];
            dst[0] = v.x; dst[1] = v.y; dst[2] = v.z; dst[3] = v.w;
        }
    }
#endif
    __syncthreads();

    const int lane = tid & 31;
    const int wave = tid >> 5;
    const int n0   = wave * 16;       // wave's N tile
    const int mrow = lane & 15;       // A: M = lane % 16
    const int g    = lane >> 4;       // lane group (0: K 0..1, 1: K 2..3)
    const int ncol = lane & 15;       // B/D: N = lane % 16

    v8f acc = {};

    // A fragment source: As[mrow][k + 2g + {0,1}]  (ds_load_b64, conflict-free)
    const float* arow = &As[mrow * LDA + 2 * g];
    // B fragment source: W[(k + 2g + {0,1}) * OUT_F + n0 + ncol]
    const float* bcol = W + n0 + ncol;

    #pragma unroll 4
    for (int k = 0; k < IN_F; k += 4) {
        v2f a, b;
        a[0] = arow[k];
        a[1] = arow[k + 1];
        b[0] = bcol[(k + 2 * g) * OUT_F];
        b[1] = bcol[(k + 2 * g + 1) * OUT_F];
        // 8 args: (neg_a, A, neg_b, B, c_mod, C, reuse_a, reuse_b)
        acc = __builtin_amdgcn_wmma_f32_16x16x4_f32(
            false, a, false, b, (short)0, acc, false, false);
    }

    // D layout: VGPR r -> M = r + 8*g, N = ncol.
    float* out = S + (size_t)m0 * OUT_F + n0 + ncol;
    #pragma unroll
    for (int r = 0; r < 8; ++r) {
        out[(r + 8 * g) * OUT_F] = acc[r];
    }
}

// --- Kernel 2: out[n, :] = bias[:]  (float4 broadcast) ----------------------
__global__ void __launch_bounds__(256)
init_bias(float* __restrict__ out, const float* __restrict__ bias, int n4) {
    int idx = blockIdx.x * 256 + threadIdx.x;
    if (idx < n4) {
        // OUT_F/4 == 32 float4 groups per row; idx%32 == lane id here.
        float4 b = ((const float4*)bias)[idx & 31];
        ((float4*)out)[idx] = b;
    }
}

// --- Kernel 3: COO SpMM scatter-add, one wave per edge ----------------------
// Each lane handles 4 consecutive features (32 lanes x float4 = 128 features).
__global__ void __launch_bounds__(256)
spmm_atomic(const int* __restrict__ rows, const int* __restrict__ cols,
            const float* __restrict__ vals, const float* __restrict__ support,
            float* __restrict__ out, int n_edges) {
    int gid  = blockIdx.x * 256 + threadIdx.x;
    int eid  = gid >> 5;
    int lane = gid & 31;
    if (eid >= n_edges) return;

    int   r = rows[eid];
    int   c = cols[eid];
    float v = vals[eid];

    float4 m = ((const float4*)(support + (size_t)c * OUT_F))[lane];

    float* dst = out + (size_t)r * OUT_F + lane * 4;
    unsafeAtomicAdd(dst + 0, m.x * v);   // global_atomic_add_f32
    unsafeAtomicAdd(dst + 1, m.y * v);
    unsafeAtomicAdd(dst + 2, m.z * v);
    unsafeAtomicAdd(dst + 3, m.w * v);
}

// ---------------------------------------------------------------------------
extern "C" void kernel_launch(void* const* d_in, const int* in_sizes, int n_in,
                              void* d_out, int out_size, void* d_ws, size_t ws_size,
                              hipStream_t stream) {
    const float* X    = (const float*)d_in[0];
    const int*   arow = (const int*)  d_in[1];
    const int*   acol = (const int*)  d_in[2];
    const float* aval = (const float*)d_in[3];
    const float* W    = (const float*)d_in[4];
    const float* bias = (const float*)d_in[5];
    float*       out  = (float*)d_out;
    float*       supp = (float*)d_ws;          // [n_nodes x 128] = 51.2 MB

    const int n_nodes = in_sizes[0] / IN_F;    // 100000 (divisible by 16)
    const int n_edges = in_sizes[1];           // 1,600,000

    // 1) support = X @ W
    gemm_wmma<<<n_nodes / 16, 256, 0, stream>>>(X, W, supp, n_nodes);

    // 2) out = broadcast(bias)
    const int n4 = n_nodes * (OUT_F / 4);
    init_bias<<<(n4 + 255) / 256, 256, 0, stream>>>(out, bias, n4);

    // 3) out[row] += val * support[col]
    spmm_atomic<<<(n_edges + 7) / 8, 256, 0, stream>>>(
        arow, acol, aval, supp, out, n_edges);
}